// SparseMaxPool_32074815766744
// MI455X (gfx1250) — compile-verified
//
#include <hip/hip_runtime.h>

#define NN 128
#define TILE_ELEMS (NN * NN)   // 16384 floats = 64 KB

// One block per (b,d) row. Build the 128x128 tile in LDS, then stream it to
// global memory with CDNA5 async LDS->memory stores (tracked by ASYNCcnt).
__global__ __launch_bounds__(256) void sparse_maxpool_kernel(
    const float* __restrict__ x, float* __restrict__ out) {
  __shared__ __align__(16) float tile[TILE_ELEMS];
  const int row = blockIdx.x;   // b*D + d
  const int tid = threadIdx.x;  // 0..255

  // ---- 1) zero the 64KB tile (ds_store_b128) ----
  float4* t4 = reinterpret_cast<float4*>(tile);
  const float4 z = make_float4(0.f, 0.f, 0.f, 0.f);
#pragma unroll
  for (int i = 0; i < TILE_ELEMS / 4 / 256; ++i) t4[tid + i * 256] = z;
  __syncthreads();

  // ---- 2) main diagonal = x row ----
  const float* xr = x + (size_t)row * NN;
  if (tid < NN) tile[tid * (NN + 1)] = xr[tid];
  __syncthreads();

  // ---- 3) max-pool cascade, written in place onto sparse diagonals ----
  // Band for step with (stride, offset): element r lives at
  //   tile[(r*stride)*128 + (r*stride + offset)] = tile[r*stride*129 + offset]
  int stride = 1, offset = 0;
  int prev_stride = 1, prev_off = 0;  // diagonal: stride 1, offset 0
  int len = NN;
  const int counts[4] = {15, 8, 8, 8};
#pragma unroll 1
  for (int ci = 0; ci < 4; ++ci) {
#pragma unroll 1
    for (int k = 0; k < counts[ci]; ++k) {
      offset += stride;
      const int kern = (ci > 0 && k == 0) ? 3 : 2;
      const int pstr = (kern == 3) ? 2 : 1;
      const int outlen = (len - kern) / pstr + 1;
      if (tid < outlen) {
        const int base = tid * pstr;
        float v0 = tile[(base    ) * prev_stride * (NN + 1) + prev_off];
        float v1 = tile[(base + 1) * prev_stride * (NN + 1) + prev_off];
        float v = fmaxf(v0, v1);
        if (kern == 3) {
          float v2 = tile[(base + 2) * prev_stride * (NN + 1) + prev_off];
          v = fmaxf(v, v2);
        }
        tile[tid * stride * (NN + 1) + offset] = v;
      }
      __syncthreads();  // band complete before next step reads it
      prev_stride = stride;
      prev_off = offset;
      len = outlen;
    }
    stride <<= 1;
  }

  // ---- 4) stream tile -> global via async LDS->memory b128 stores ----
  // Low 32 bits of the generic LDS pointer are the workgroup-relative LDS
  // byte offset (aperture bits live in [63:32]).
  const unsigned lds_base = (unsigned)(unsigned long long)(const void*)tile;
  const unsigned long long gbase =
      (unsigned long long)out + (unsigned long long)row * (TILE_ELEMS * 4ull);
#pragma unroll
  for (int i = 0; i < TILE_ELEMS * 4 / 16 / 256; ++i) {  // 16 iters
    const unsigned boff = (unsigned)(tid + i * 256) * 16u;
    unsigned long long ga = gbase + boff;
    unsigned la = lds_base + boff;
    asm volatile("global_store_async_from_lds_b128 %0, %1, off"
                 :
                 : "v"(ga), "v"(la)
                 : "memory");
  }
  // Ensure all async stores retire before the wave ends.
  asm volatile("s_wait_asynccnt 0" ::: "memory");
}

extern "C" void kernel_launch(void* const* d_in, const int* in_sizes, int n_in,
                              void* d_out, int out_size, void* d_ws, size_t ws_size,
                              hipStream_t stream) {
  const float* x = (const float*)d_in[0];
  float* out = (float*)d_out;
  const int rows = in_sizes[0] / NN;  // B*D = 4096
  sparse_maxpool_kernel<<<rows, 256, 0, stream>>>(x, out);
}